// GraphConvolution_73950746902582
// MI455X (gfx1250) — compile-verified
//
#include <hip/hip_runtime.h>
#include <math.h>

#define D 128

typedef __attribute__((ext_vector_type(2))) float v2f;
typedef __attribute__((ext_vector_type(8))) float v8f;

// ---------------- zero workspace (h_acc [n*D] floats + deg [n] ints, contiguous) ----------------
__global__ void gcn2_zero_kernel(float* __restrict__ p, long long n) {
    long long i = (long long)blockIdx.x * blockDim.x + threadIdx.x;
    long long stride = (long long)gridDim.x * blockDim.x;
    for (; i < n; i += stride) p[i] = 0.0f;
}

// ---------------- in-degree via int atomics ----------------
__global__ void gcn2_deg_kernel(const int* __restrict__ edst, int* __restrict__ deg, int E) {
    int e = blockIdx.x * blockDim.x + threadIdx.x;
    if (e < E) atomicAdd(&deg[edst[e]], 1);
}

// ---------------- dinv = rsqrt(max(deg,1)) ----------------
__global__ void gcn2_dinv_kernel(const int* __restrict__ deg, float* __restrict__ dinv, int n) {
    int i = blockIdx.x * blockDim.x + threadIdx.x;
    if (i < n) {
        int d = deg[i];
        if (d < 1) d = 1;
        dinv[i] = rsqrtf((float)d);
    }
}

// ---------------- SpMM: one wave per edge, float4 per lane, f32 global atomics ----------------
__global__ void gcn2_spmm_kernel(const float* __restrict__ x,
                                 const int* __restrict__ esrc,
                                 const int* __restrict__ edst,
                                 const float* __restrict__ dinv,
                                 float* __restrict__ hacc, int E) {
    long long tid = (long long)blockIdx.x * blockDim.x + threadIdx.x;
    int e = (int)(tid >> 5);          // one wave (32 lanes) per edge
    if (e >= E) return;
    int c = ((int)tid & 31) * 4;      // 32 lanes x 4 floats = 128 features
    int s = esrc[e];
    int d = edst[e];
    float sc = dinv[s];
    const float4 v = *(const float4*)(x + (long long)s * D + c);
    float* o = hacc + (long long)d * D + c;
    atomicAdd(o + 0, v.x * sc);
    atomicAdd(o + 1, v.y * sc);
    atomicAdd(o + 2, v.z * sc);
    atomicAdd(o + 3, v.w * sc);
}

// ---------------- fused: support = (1-a)*hacc*dinv + a*h0 ; out = th*(support@W) + (1-th)*support + x
// block = 256 threads = 8 waves; block handles 16 rows, each wave one 16-col tile.
__global__ __launch_bounds__(256) void gcn2_gemm_kernel(
    const float* __restrict__ x,      // inputs [n,D]
    const float* __restrict__ h0,     // [n,D]
    const float* __restrict__ W,      // [D,D] row-major
    const float* __restrict__ hacc,   // [n,D]
    const float* __restrict__ dinv,   // [n]
    const float* __restrict__ lam_p,
    const float* __restrict__ alp_p,
    const int*   __restrict__ l_p,
    float* __restrict__ out, int n)
{
    __shared__ float sA[16 * D];      // support tile, 8 KB

    const float alpha = alp_p[0];
    const float theta = logf(lam_p[0] / (float)l_p[0] + 1.0f);
    const int r0 = blockIdx.x * 16;

    // stage support tile into LDS (coalesced: 256 threads x 8 elems)
    for (int idx = threadIdx.x; idx < 16 * D; idx += 256) {
        int r = idx >> 7;             // / D
        int c = idx & (D - 1);
        int row = r0 + r;
        float s = 0.0f;
        if (row < n) {
            long long g = (long long)row * D + c;
            s = (1.0f - alpha) * hacc[g] * dinv[row] + alpha * h0[g];
        }
        sA[idx] = s;
    }
    __syncthreads();

    const int lane  = threadIdx.x & 31;
    const int ntile = threadIdx.x >> 5;   // 0..7 -> column tile
    const int n0    = ntile * 16;
    const int mrow  = lane & 15;          // M for A-frag, N for B/C frags
    const int khalf = lane >> 4;          // 0: K={0,1}, 1: K={2,3}

    v8f acc = {};
    #pragma unroll
    for (int k0 = 0; k0 < D; k0 += 4) {
        int kk = k0 + khalf * 2;
        // A 16x4 f32 fragment: lane holds A[mrow][kk], A[mrow][kk+1]  (8B-aligned LDS read)
        v2f a = *(const v2f*)&sA[mrow * D + kk];
        // B 4x16 f32 fragment: lane holds W[kk][n0+mrow], W[kk+1][n0+mrow]
        v2f b;
        b.x = W[(kk + 0) * D + n0 + mrow];
        b.y = W[(kk + 1) * D + n0 + mrow];
        acc = __builtin_amdgcn_wmma_f32_16x16x4_f32(
            /*neg_a=*/false, a, /*neg_b=*/false, b,
            /*c_mod=*/(short)0, acc, /*reuse_a=*/false, /*reuse_b=*/false);
    }

    // C/D layout: VGPR r -> M = r + 8*khalf, N = mrow (within tile)
    #pragma unroll
    for (int r = 0; r < 8; ++r) {
        int M = r + khalf * 8;
        int row = r0 + M;
        if (row < n) {
            int col = n0 + mrow;
            long long g = (long long)row * D + col;
            float sup = sA[M * D + col];
            out[g] = theta * acc[r] + (1.0f - theta) * sup + x[g];
        }
    }
}

extern "C" void kernel_launch(void* const* d_in, const int* in_sizes, int n_in,
                              void* d_out, int out_size, void* d_ws, size_t ws_size,
                              hipStream_t stream) {
    (void)n_in; (void)out_size; (void)ws_size;
    const float* x    = (const float*)d_in[0];
    const int*   esrc = (const int*)d_in[1];
    const int*   edst = (const int*)d_in[2];
    const float* h0   = (const float*)d_in[3];
    const float* W    = (const float*)d_in[4];
    const float* lam  = (const float*)d_in[5];
    const float* alp  = (const float*)d_in[6];
    const int*   lp   = (const int*)d_in[7];
    float* out = (float*)d_out;

    const int n = in_sizes[0] / D;
    const int E = in_sizes[1];

    // workspace layout: h_acc [n*D] f32 | deg [n] i32 | dinv [n] f32
    float* hacc = (float*)d_ws;
    int*   deg  = (int*)(hacc + (size_t)n * D);
    float* dinv = (float*)(deg + n);

    const long long zn = (long long)n * D + n;   // zero hacc + deg (contiguous)
    int zblocks = (int)((zn + 255) / 256);
    gcn2_zero_kernel<<<zblocks, 256, 0, stream>>>((float*)d_ws, zn);

    gcn2_deg_kernel<<<(E + 255) / 256, 256, 0, stream>>>(edst, deg, E);
    gcn2_dinv_kernel<<<(n + 255) / 256, 256, 0, stream>>>(deg, dinv, n);

    const long long sthreads = (long long)E * 32;   // one wave per edge
    gcn2_spmm_kernel<<<(int)((sthreads + 255) / 256), 256, 0, stream>>>(x, esrc, edst, dinv, hacc, E);

    gcn2_gemm_kernel<<<(n + 15) / 16, 256, 0, stream>>>(x, h0, W, hacc, dinv, lam, alp, lp, out, n);
}